// DenseInterTripletLoss_69672959475800
// MI455X (gfx1250) — compile-verified
//
#include <hip/hip_runtime.h>

#define GS 8
#define BATCH 2
#define CH 64
#define HH 480
#define WW 640
#define HC 60
#define WC 80
#define NN (HC*WC)        // 4800
#define NTILE (NN/16)     // 300
#define NMASK ((NTILE+31)/32)  // 10
#define EPSF 1e-8f

typedef float v2f __attribute__((ext_vector_type(2)));
typedef float v8f __attribute__((ext_vector_type(8)));

// ---------------------------------------------------------------- init
__global__ void k_init(float* accum) {
    if (threadIdx.x == 0) { accum[0] = 0.f; accum[1] = 0.f; }
}

// ---------------------------------------------------------------- prep:
// warp grid centers via homo12, bilinear-sample desc2 -> pos_sim,
// match_mask, 4-nearest neighbor column ids.
__global__ void k_prep(const float* __restrict__ desc1,
                       const float* __restrict__ desc2,
                       const float* __restrict__ homo12,
                       float* __restrict__ pos, float* __restrict__ mmask,
                       int* __restrict__ neigh, float* __restrict__ accum)
{
    int t = blockIdx.x * blockDim.x + threadIdx.x;
    if (t >= BATCH * NN) return;
    int b = t / NN, n = t % NN;
    int cy = n / WC, cx = n % WC;
    float x = cx * (float)GS + GS * 0.5f - 0.5f;
    float y = cy * (float)GS + GS * 0.5f - 0.5f;
    const float* Hm = homo12 + b * 9;
    float qx = Hm[0]*x + Hm[1]*y + Hm[2];
    float qy = Hm[3]*x + Hm[4]*y + Hm[5];
    float qz = Hm[6]*x + Hm[7]*y + Hm[8];
    float wx = qx / (qz + EPSF), wy = qy / (qz + EPSF);

    float mk = (wy >= 0.f && wy <= (float)(HH-1) &&
                wx >= 0.f && wx <= (float)(WW-1)) ? 1.f : 0.f;
    mmask[t] = mk;
    atomicAdd(&accum[1], mk);

    // bilinear sample of desc2 at grid coords
    float dy = (wy - 3.5f) / 8.f, dx = (wx - 3.5f) / 8.f;
    float y0 = floorf(dy), x0 = floorf(dx);
    float fy = dy - y0, fx = dx - x0;
    float wgt[4] = { (1.f-fy)*(1.f-fx), (1.f-fy)*fx, fy*(1.f-fx), fy*fx };
    float cyf[4] = { y0, y0, y0+1.f, y0+1.f };
    float cxf[4] = { x0, x0+1.f, x0, x0+1.f };
    int idx4[4]; float wv[4];
    #pragma unroll
    for (int k = 0; k < 4; k++) {
        bool valid = (cyf[k] >= 0.f) && (cyf[k] <= (float)(HC-1)) &&
                     (cxf[k] >= 0.f) && (cxf[k] <= (float)(WC-1));
        int yc = (int)fminf(fmaxf(cyf[k], 0.f), (float)(HC-1));
        int xc = (int)fminf(fmaxf(cxf[k], 0.f), (float)(WC-1));
        idx4[k] = yc * WC + xc;
        wv[k] = valid ? wgt[k] : 0.f;
    }
    const float* D1 = desc1 + (size_t)b * CH * NN;
    const float* D2 = desc2 + (size_t)b * CH * NN;
    float dot = 0.f, nrm = 0.f;
    for (int ch = 0; ch < CH; ch++) {
        const float* p = D2 + ch * NN;
        float v = wv[0]*p[idx4[0]] + wv[1]*p[idx4[1]] +
                  wv[2]*p[idx4[2]] + wv[3]*p[idx4[3]];
        dot += v * D1[ch * NN + n];
        nrm += v * v;
    }
    float ps = dot / (sqrtf(nrm) + EPSF);
    pos[t] = sqrtf(fmaxf(2.f - 2.f * ps, EPSF));

    // 4 nearest grid centers to (wy,wx): search a clamped 4x4 lattice window
    int sy = (int)floorf(dy) - 1; sy = sy < 0 ? 0 : (sy > HC-4 ? HC-4 : sy);
    int sx = (int)floorf(dx) - 1; sx = sx < 0 ? 0 : (sx > WC-4 ? WC-4 : sx);
    float cd[16]; int cid[16];
    #pragma unroll
    for (int jy = 0; jy < 4; jy++)
        #pragma unroll
        for (int jx = 0; jx < 4; jx++) {
            int iy = sy + jy, ix = sx + jx;
            float ddy = wy - (iy * 8.f + 3.5f);
            float ddx = wx - (ix * 8.f + 3.5f);
            cd[jy*4+jx] = ddy*ddy + ddx*ddx;
            cid[jy*4+jx] = iy * WC + ix;
        }
    int out4[4];
    #pragma unroll
    for (int pick = 0; pick < 4; pick++) {
        int bi = 0; float bd = cd[0]; int bv = cid[0];
        #pragma unroll
        for (int q = 1; q < 16; q++) {
            if (cd[q] < bd || (cd[q] == bd && cid[q] < bv)) {
                bd = cd[q]; bv = cid[q]; bi = q;
            }
        }
        out4[pick] = bv; cd[bi] = 3.0e38f;
    }
    neigh[t*4+0] = out4[0]; neigh[t*4+1] = out4[1];
    neigh[t*4+2] = out4[2]; neigh[t*4+3] = out4[3];
}

// ---------------------------------------------------------------- visibility:
// per cell, product over its 64 pixels of (bilinear(ones, warp(homo21)) > 0)
__global__ void k_vis(const float* __restrict__ homo21,
                      float* __restrict__ colpen)
{
    int t = blockIdx.x * blockDim.x + threadIdx.x;
    if (t >= BATCH * NN) return;
    int b = t / NN, n = t % NN;
    int cy = n / WC, cx = n % WC;
    const float* Hm = homo21 + b * 9;
    float prod = 1.f;
    for (int j = 0; j < GS; j++) {
        for (int i = 0; i < GS; i++) {
            float x = (float)(cx * GS + i);
            float y = (float)(cy * GS + j);
            float qx = Hm[0]*x + Hm[1]*y + Hm[2];
            float qy = Hm[3]*x + Hm[4]*y + Hm[5];
            float qz = Hm[6]*x + Hm[7]*y + Hm[8];
            float X = qx / (qz + EPSF), Y = qy / (qz + EPSF);
            float y0 = floorf(Y), x0 = floorf(X);
            float fy = Y - y0, fx = X - x0;
            float v = 0.f;
            v += ((y0 >= 0.f) && (y0 <= (float)(HH-1)) &&
                  (x0 >= 0.f) && (x0 <= (float)(WW-1))) ? (1.f-fy)*(1.f-fx) : 0.f;
            v += ((y0 >= 0.f) && (y0 <= (float)(HH-1)) &&
                  (x0+1.f >= 0.f) && (x0+1.f <= (float)(WW-1))) ? (1.f-fy)*fx : 0.f;
            v += ((y0+1.f >= 0.f) && (y0+1.f <= (float)(HH-1)) &&
                  (x0 >= 0.f) && (x0 <= (float)(WW-1))) ? fy*(1.f-fx) : 0.f;
            v += ((y0+1.f >= 0.f) && (y0+1.f <= (float)(HH-1)) &&
                  (x0+1.f >= 0.f) && (x0+1.f <= (float)(WW-1))) ? fy*fx : 0.f;
            prod *= (v > 0.f) ? 1.f : 0.f;
        }
    }
    colpen[t] = 5.f * (1.f - prod);
}

// ---------------------------------------------------------------- main:
// fused f32 WMMA GEMM (16x16 tile, K=64 via 16x chained 16x16x4) +
// sqrt/penalty transform + row-min, loss accumulation.
// Neighbor-penalty compares are guarded by a per-block 300-bit tile mask;
// B fragments are double-buffered so the epilogue overlaps next-tile loads.
__global__ __launch_bounds__(128) void k_main(
    const float* __restrict__ desc1, const float* __restrict__ desc2,
    const float* __restrict__ colpen, const int* __restrict__ neigh,
    const float* __restrict__ pos, const float* __restrict__ mmask,
    float* __restrict__ accum)
{
    __shared__ float red[4][16];
    __shared__ unsigned tmask[NMASK];   // bit per column tile: contains a neighbor id

    int b = blockIdx.x / NTILE;
    int rowTile = blockIdx.x % NTILE;
    int rowbase = rowTile * 16;
    int lane = threadIdx.x & 31;
    int wid  = threadIdx.x >> 5;
    int l16  = lane & 15;
    int khalf = (lane >> 4) * 2;      // lanes 0-15: K 0,1 ; lanes 16-31: K 2,3
    int rh = (lane >> 4) * 8;         // accumulator VGPR r -> row r (+8 upper half)

    const float* D1 = desc1 + (size_t)b * CH * NN;
    const float* D2 = desc2 + (size_t)b * CH * NN;

    // build the neighbor tile mask (shared by all 4 waves: same 16 rows)
    if (threadIdx.x < NMASK) tmask[threadIdx.x] = 0u;
    __syncthreads();
    if (threadIdx.x < 16) {
        int4 g = ((const int4*)neigh)[b * NN + rowbase + threadIdx.x];
        atomicOr(&tmask[(unsigned)(g.x >> 4) >> 5], 1u << ((g.x >> 4) & 31));
        atomicOr(&tmask[(unsigned)(g.y >> 4) >> 5], 1u << ((g.y >> 4) & 31));
        atomicOr(&tmask[(unsigned)(g.z >> 4) >> 5], 1u << ((g.z >> 4) & 31));
        atomicOr(&tmask[(unsigned)(g.w >> 4) >> 5], 1u << ((g.w >> 4) & 31));
    }

    // A fragments: 16 K-chunks of 4 for the 16 rows of this block
    int mrow = rowbase + l16;
    v2f afr[16];
    #pragma unroll
    for (int kk = 0; kk < 16; kk++) {
        int k0 = kk * 4 + khalf;
        v2f a; a.x = D1[k0 * NN + mrow]; a.y = D1[(k0+1) * NN + mrow];
        afr[kk] = a;
    }
    // neighbor ids for each row this lane's accumulator entries cover
    int4 ng[8];
    #pragma unroll
    for (int r = 0; r < 8; r++) {
        int rowr = rowbase + r + rh;
        ng[r] = ((const int4*)neigh)[b * NN + rowr];
    }
    __syncthreads();

    float minv[8];
    #pragma unroll
    for (int r = 0; r < 8; r++) minv[r] = 3.0e38f;

    // 300 column tiles split evenly over the 4 waves (75 each, no divergence).
    // Double-buffer B fragments: prefetch tile ct+4 while finishing tile ct.
    v2f bfr[16];
    {
        int ccol = wid * 16 + l16;
        #pragma unroll
        for (int kk = 0; kk < 16; kk++) {
            int k0 = kk * 4 + khalf;
            v2f x; x.x = D2[k0 * NN + ccol]; x.y = D2[(k0+1) * NN + ccol];
            bfr[kk] = x;
        }
    }
    for (int ct = wid; ct < NTILE; ct += 4) {
        int ccol = ct * 16 + l16;
        int ctn = ct + 4;
        v2f bnx[16];
        if (ctn < NTILE) {              // wave-uniform branch
            int ncol = ctn * 16 + l16;
            #pragma unroll
            for (int kk = 0; kk < 16; kk++) {
                int k0 = kk * 4 + khalf;
                v2f x; x.x = D2[k0 * NN + ncol]; x.y = D2[(k0+1) * NN + ncol];
                bnx[kk] = x;
            }
        }
        v8f acc = {0.f,0.f,0.f,0.f,0.f,0.f,0.f,0.f};
        #pragma unroll
        for (int kk = 0; kk < 16; kk++) {
            acc = __builtin_amdgcn_wmma_f32_16x16x4_f32(
                false, afr[kk], false, bfr[kk], (short)0, acc, false, false);
        }
        float cp = colpen[b * NN + ccol];
        bool chk = (tmask[(unsigned)ct >> 5] >> (ct & 31)) & 1u;  // wave-uniform
        #pragma unroll
        for (int r = 0; r < 8; r++) {
            float c = acc[r];
            float val = __builtin_amdgcn_sqrtf(
                            fmaxf(fmaf(c, -2.f, 2.f), EPSF)) + cp;
            if (chk) {
                int4 g = ng[r];
                if (ccol == g.x || ccol == g.y || ccol == g.z || ccol == g.w)
                    val += 5.f;
            }
            minv[r] = fminf(minv[r], val);
        }
        if (ctn < NTILE) {
            #pragma unroll
            for (int kk = 0; kk < 16; kk++) bfr[kk] = bnx[kk];
        }
    }
    // min across the 16 lanes of each half (N dimension)
    #pragma unroll
    for (int r = 0; r < 8; r++) {
        float v = minv[r];
        v = fminf(v, __shfl_xor(v, 1, 32));
        v = fminf(v, __shfl_xor(v, 2, 32));
        v = fminf(v, __shfl_xor(v, 4, 32));
        v = fminf(v, __shfl_xor(v, 8, 32));
        minv[r] = v;
    }
    if (l16 == 0) {   // lane 0 -> rows 0..7, lane 16 -> rows 8..15
        #pragma unroll
        for (int r = 0; r < 8; r++) red[wid][rh + r] = minv[r];
    }
    __syncthreads();
    if (threadIdx.x < 16) {
        int rl = threadIdx.x;
        float m = fminf(fminf(red[0][rl], red[1][rl]),
                        fminf(red[2][rl], red[3][rl]));
        int row = rowbase + rl;
        float ps = pos[b * NN + row];
        float mk = mmask[b * NN + row];
        float l = fmaxf(ps - m + 1.0f, 0.f);
        float contrib = l * l * mk;
        contrib += __shfl_xor(contrib, 1, 32);
        contrib += __shfl_xor(contrib, 2, 32);
        contrib += __shfl_xor(contrib, 4, 32);
        contrib += __shfl_xor(contrib, 8, 32);
        if (rl == 0) atomicAdd(&accum[0], contrib);
    }
}

// ---------------------------------------------------------------- final
__global__ void k_final(const float* __restrict__ accum, float* __restrict__ out) {
    if (threadIdx.x == 0) out[0] = accum[0] / accum[1];
}

extern "C" void kernel_launch(void* const* d_in, const int* in_sizes, int n_in,
                              void* d_out, int out_size, void* d_ws, size_t ws_size,
                              hipStream_t stream)
{
    (void)in_sizes; (void)n_in; (void)out_size; (void)ws_size;
    const float* desc1  = (const float*)d_in[2];
    const float* desc2  = (const float*)d_in[3];
    const float* homo12 = (const float*)d_in[4];
    const float* homo21 = (const float*)d_in[5];

    float* ws     = (float*)d_ws;
    float* pos    = ws;                       // B*N floats
    float* mmask  = ws + BATCH * NN;          // B*N floats
    float* colpen = ws + 2 * BATCH * NN;      // B*N floats
    int*   neigh  = (int*)(ws + 3 * BATCH * NN);          // 4*B*N ints (16B aligned)
    float* accum  = ws + 3 * BATCH * NN + 4 * BATCH * NN; // 2 floats

    int tot = BATCH * NN;
    int blk = 128;
    int nb = (tot + blk - 1) / blk;

    k_init<<<1, 32, 0, stream>>>(accum);
    k_prep<<<nb, blk, 0, stream>>>(desc1, desc2, homo12, pos, mmask, neigh, accum);
    k_vis <<<nb, blk, 0, stream>>>(homo21, colpen);
    k_main<<<BATCH * NTILE, 128, 0, stream>>>(desc1, desc2, colpen, neigh,
                                              pos, mmask, accum);
    k_final<<<1, 32, 0, stream>>>(accum, (float*)d_out);
}